// CPAM_fuse_50732153700481
// MI455X (gfx1250) — compile-verified
//
#include <hip/hip_runtime.h>

typedef float v2f __attribute__((ext_vector_type(2)));
typedef float v8f __attribute__((ext_vector_type(8)));

#define CCH   16
#define HW    784
#define N_TOT 12544   // C*H*W
#define TPB   256
#define ROWS_PER_BLOCK 128

// ---------------------------------------------------------------------------
// Kernel 1: the three 1x1 convs as WMMA f32 GEMMs.
// One wave (32 threads) per 16-column spatial tile; for each of wq/wk/wv we
// accumulate D = W[16x16] * X[16x16tile] + bias via 4 chained
// V_WMMA_F32_16X16X4_F32 ops (K split 16 = 4 x 4).
//
// ISA VGPR layouts (CDNA5 ISA):
//   A 16x4 f32 : lane L -> M = L%16 ; VGPR j -> K = (L<16 ? j : 2+j)
//   B 4x16 f32 : lane L -> N = L%16 ; VGPR j -> K = (L<16 ? j : 2+j)
//   C/D 16x16  : lane L -> N = L%16 ; VGPR r -> M = r + (L<16 ? 0 : 8)
// ---------------------------------------------------------------------------
__global__ __launch_bounds__(32) void proj_wmma(
    const float* __restrict__ x,
    const float* __restrict__ wq, const float* __restrict__ bq,
    const float* __restrict__ wk, const float* __restrict__ bk,
    const float* __restrict__ wv, const float* __restrict__ bv,
    float* __restrict__ q, float* __restrict__ k, float* __restrict__ v)
{
    const int lane = threadIdx.x;            // 0..31
    const int p    = blockIdx.x * 16 + (lane & 15);  // spatial column
    const int mrow = lane & 15;              // M index for A loads
    const int koff = (lane >> 4) * 2;        // K-half offset for A/B loads
    const int moff = (lane >> 4) * 8;        // M-half offset for C/D

    // B-tile of x is shared by all three projections: preload once.
    v2f xb[4];
#pragma unroll
    for (int kk = 0; kk < 4; ++kk) {
        xb[kk].x = x[(4 * kk + koff + 0) * HW + p];
        xb[kk].y = x[(4 * kk + koff + 1) * HW + p];
    }

    const float* W[3]   = {wq, wk, wv};
    const float* Bias[3] = {bq, bk, bv};
    float*       Out[3] = {q, k, v};

#pragma unroll
    for (int t = 0; t < 3; ++t) {
        const float* w = W[t];
        const float* b = Bias[t];

        v8f c;
#pragma unroll
        for (int r = 0; r < 8; ++r) c[r] = b[moff + r];   // bias as C init

#pragma unroll
        for (int kk = 0; kk < 4; ++kk) {
            v2f a;
            a.x = w[mrow * CCH + 4 * kk + koff + 0];
            a.y = w[mrow * CCH + 4 * kk + koff + 1];
            c = __builtin_amdgcn_wmma_f32_16x16x4_f32(
                    /*neg_a=*/false, a, /*neg_b=*/false, xb[kk],
                    /*c_mod=*/(short)0, c, /*reuse_a=*/false, /*reuse_b=*/false);
        }

        float* o = Out[t];
#pragma unroll
        for (int r = 0; r < 8; ++r) o[(moff + r) * HW + p] = c[r];
    }
}

// ---------------------------------------------------------------------------
// Order-preserving float <-> uint encode for atomic min/max of floats.
// ---------------------------------------------------------------------------
__device__ inline unsigned fenc(float f) {
    unsigned u = __float_as_uint(f);
    return (u & 0x80000000u) ? ~u : (u | 0x80000000u);
}
__device__ inline float fdec(unsigned u) {
    unsigned b = (u & 0x80000000u) ? (u & 0x7FFFFFFFu) : ~u;
    return __uint_as_float(b);
}

__global__ void init_minmax(unsigned* mm) { mm[0] = 0u; mm[1] = 0u; }

// Kernel 2: global max(k) and min(k) (min tracked as max(-k)).
__global__ __launch_bounds__(TPB) void k_minmax(const float* __restrict__ k,
                                                unsigned* __restrict__ mm)
{
    int i = blockIdx.x * blockDim.x + threadIdx.x;
    unsigned mx = 0u, mn = 0u;
    if (i < N_TOT) {
        float f = k[i];
        mx = fenc(f);
        mn = fenc(-f);
    }
#pragma unroll
    for (int off = 16; off; off >>= 1) {
        unsigned omx = (unsigned)__shfl_xor((int)mx, off, 32);
        unsigned omn = (unsigned)__shfl_xor((int)mn, off, 32);
        mx = mx > omx ? mx : omx;
        mn = mn > omn ? mn : omn;
    }
    if ((threadIdx.x & 31) == 0) {
        atomicMax(&mm[0], mx);
        atomicMax(&mm[1], mn);
    }
}

// ---------------------------------------------------------------------------
// Kernel 3: out[m] = sum_n v[n]*exp(q[m]k[n]-c_m) / sum_n exp(q[m]k[n]-c_m)
// c_m = exact row max = q[m]>=0 ? q[m]*kmax : q[m]*kmin, so args <= 0 and at
// least one term is exp(0)=1 -> no overflow, denominator >= 1.
// log2(e) folded into s and c so the inner loop is 1 fma + 1 v_exp_f32 + 1 fma.
// k cached in LDS (50 KB); v streams from L2 (fully resident).
// ---------------------------------------------------------------------------
__global__ __launch_bounds__(TPB) void attn_rows(
    const float* __restrict__ q, const float* __restrict__ kk,
    const float* __restrict__ vv, const unsigned* __restrict__ mm,
    float* __restrict__ out)
{
    __shared__ float lds_k[N_TOT];
    __shared__ float red[2][TPB / 32];

    const int tid = threadIdx.x;
    for (int i = tid; i < N_TOT; i += TPB) lds_k[i] = kk[i];
    __syncthreads();

    const float kmax = fdec(mm[0]);
    const float kmin = -fdec(mm[1]);
    const float L2E  = 1.4426950408889634f;
    const int row0   = blockIdx.x * ROWS_PER_BLOCK;

    for (int rr = 0; rr < ROWS_PER_BLOCK; ++rr) {
        const int   m   = row0 + rr;
        const float s   = q[m];
        const float cmx = (s >= 0.0f) ? s * kmax : s * kmin;
        const float s2  = s * L2E;
        const float c2  = cmx * L2E;

        float dacc = 0.0f, uacc = 0.0f;
        for (int n = tid; n < N_TOT; n += TPB) {
            float e = __builtin_amdgcn_exp2f(__builtin_fmaf(s2, lds_k[n], -c2));
            dacc += e;
            uacc = __builtin_fmaf(vv[n], e, uacc);
        }

#pragma unroll
        for (int off = 16; off; off >>= 1) {
            dacc += __shfl_xor(dacc, off, 32);
            uacc += __shfl_xor(uacc, off, 32);
        }
        const int wid = tid >> 5;
        if ((tid & 31) == 0) { red[0][wid] = dacc; red[1][wid] = uacc; }
        __syncthreads();
        if (tid == 0) {
            float D = 0.0f, U = 0.0f;
#pragma unroll
            for (int w = 0; w < TPB / 32; ++w) { D += red[0][w]; U += red[1][w]; }
            out[m] = U / D;
        }
        __syncthreads();
    }
}

// ---------------------------------------------------------------------------
extern "C" void kernel_launch(void* const* d_in, const int* in_sizes, int n_in,
                              void* d_out, int out_size, void* d_ws, size_t ws_size,
                              hipStream_t stream)
{
    const float* x  = (const float*)d_in[0];
    const float* wq = (const float*)d_in[1];
    const float* bq = (const float*)d_in[2];
    const float* wk = (const float*)d_in[3];
    const float* bk = (const float*)d_in[4];
    const float* wv = (const float*)d_in[5];
    const float* bv = (const float*)d_in[6];
    float* out = (float*)d_out;

    float*    q  = (float*)d_ws;
    float*    k  = q + N_TOT;
    float*    v  = k + N_TOT;
    unsigned* mm = (unsigned*)(v + N_TOT);

    init_minmax<<<1, 1, 0, stream>>>(mm);
    proj_wmma<<<HW / 16, 32, 0, stream>>>(x, wq, bq, wk, bk, wv, bv, q, k, v);
    k_minmax<<<(N_TOT + TPB - 1) / TPB, TPB, 0, stream>>>(k, mm);
    attn_rows<<<N_TOT / ROWS_PER_BLOCK, TPB, 0, stream>>>(q, k, v, mm, out);
}